// FusedMoEDispatch_45775761440770
// MI455X (gfx1250) — compile-verified
//
#include <hip/hip_runtime.h>
#include <hip/hip_bf16.h>
#include <math.h>

#define NUM_EXPERTS 64
#define TOPK 2
#define ROW_FLOATS 4096
#define ROW_VEC4 (ROW_FLOATS / 4)          // 1024 16-byte vectors per row
#define EXPAND_THREADS 256
#define CHUNKS (ROW_VEC4 / EXPAND_THREADS) // 4 vectors per thread

// Native clang vector types (work with nontemporal builtins on host + device,
// and match the async-builtin pointee type on device).
typedef float v4f __attribute__((vector_size(4 * sizeof(float))));
typedef int   v4i __attribute__((vector_size(4 * sizeof(int))));

// ---- CDNA5 async global<->LDS path (gfx1250) --------------------------------
// Builtin signature (confirmed by clang diagnostic in an earlier round):
// (v4i addrspace(1)* gaddr, v4i addrspace(3)* lds, imm offset, imm cpol).
typedef __attribute__((address_space(1))) v4i as1_v4i;
typedef __attribute__((address_space(3))) v4i as3_v4i;

#if __has_builtin(__builtin_amdgcn_global_load_async_to_lds_b128) && \
    __has_builtin(__builtin_amdgcn_global_store_async_from_lds_b128)
#define USE_ASYNC_LDS 1
#else
#define USE_ASYNC_LDS 0
#endif

#if __has_builtin(__builtin_amdgcn_s_wait_asynccnt)
#define WAIT_ASYNCCNT(n) __builtin_amdgcn_s_wait_asynccnt(n)
#else
#define WAIT_ASYNCCNT(n) asm volatile("s_wait_asynccnt %0" ::"i"(n) : "memory")
#endif

// ---- Router: softmax + top-2 + renormalize ----------------------------------
// Renormalized top-2 softmax weights: the full softmax denominator cancels, so
// w1 = 1 / (1 + exp(l2 - l1)), w2 = 1 - w1. Strict '>' scan in ascending
// expert order matches jax.lax.top_k tie-breaking (lower index first).
// Branchless selects -> v_cmp + v_cndmask, no exec-mask divergence.
__global__ void moe_router_kernel(const float* __restrict__ logits,
                                  int* __restrict__ expert_idx,
                                  float* __restrict__ expert_w,
                                  int batch) {
    int t = blockIdx.x * blockDim.x + threadIdx.x;
    if (t >= batch) return;

    const v4f* lp = (const v4f*)(logits + (size_t)t * NUM_EXPERTS);
    float m1 = -3.402823466e38f, m2 = -3.402823466e38f;
    int i1 = 0, i2 = 0;
#pragma unroll
    for (int q = 0; q < NUM_EXPERTS / 4; ++q) {
        v4f v4 = lp[q];
#pragma unroll
        for (int j = 0; j < 4; ++j) {
            const float v = v4[j];
            const int e = q * 4 + j;
            const bool gt1 = v > m1;
            const bool gt2 = v > m2;
            // Compute runner-up from OLD leader before updating the leader.
            const float nm2 = gt1 ? m1 : (gt2 ? v : m2);
            const int   ni2 = gt1 ? i1 : (gt2 ? e : i2);
            m1 = gt1 ? v : m1;
            i1 = gt1 ? e : i1;
            m2 = nm2;
            i2 = ni2;
        }
    }
    float w1 = 1.0f / (1.0f + expf(m2 - m1));
    expert_idx[2 * t + 0] = i1;
    expert_idx[2 * t + 1] = i2;
    expert_w[2 * t + 0] = w1;
    expert_w[2 * t + 1] = 1.0f - w1;
}

// ---- Token expansion: out[2t] = out[2t+1] = x[t] ----------------------------
// One block per token row (16 KB). Async-load the row into LDS with 128-bit
// async ops, wait on ASYNCcnt (per-wave; each thread stores exactly what it
// loaded, so no cross-wave barrier needed), then async-store to both
// destination rows straight out of LDS. Pure HBM-bandwidth kernel:
// 768 MB total traffic -> ~33 us floor at 23.3 TB/s.
__global__ void __launch_bounds__(EXPAND_THREADS)
moe_expand_kernel(const float* __restrict__ x, float* __restrict__ out) {
    __shared__ v4f tile[ROW_VEC4]; // 16 KB
    const int tid = threadIdx.x;
    const size_t tok = blockIdx.x;

    const v4f* src = (const v4f*)x + tok * ROW_VEC4;
    v4f* dst0 = (v4f*)out + (tok * 2) * ROW_VEC4;
    v4f* dst1 = dst0 + ROW_VEC4;

#if USE_ASYNC_LDS
#pragma unroll
    for (int c = 0; c < CHUNKS; ++c) {
        int i = c * EXPAND_THREADS + tid;
        __builtin_amdgcn_global_load_async_to_lds_b128(
            (as1_v4i*)(src + i), (as3_v4i*)(&tile[i]), 0, 0);
    }
    WAIT_ASYNCCNT(0); // this wave's LDS writes have landed
#pragma unroll
    for (int c = 0; c < CHUNKS; ++c) {
        int i = c * EXPAND_THREADS + tid;
        __builtin_amdgcn_global_store_async_from_lds_b128(
            (as1_v4i*)(dst0 + i), (as3_v4i*)(&tile[i]), 0, 0);
        __builtin_amdgcn_global_store_async_from_lds_b128(
            (as1_v4i*)(dst1 + i), (as3_v4i*)(&tile[i]), 0, 0);
    }
    WAIT_ASYNCCNT(0);
#else
    // Fallback: direct 128-bit nontemporal copy (still bandwidth-optimal).
#pragma unroll
    for (int c = 0; c < CHUNKS; ++c) {
        int i = c * EXPAND_THREADS + tid;
        v4f v = __builtin_nontemporal_load(src + i);
        __builtin_nontemporal_store(v, dst0 + i);
        __builtin_nontemporal_store(v, dst1 + i);
    }
#endif
}

// ---- Host launch ------------------------------------------------------------
extern "C" void kernel_launch(void* const* d_in, const int* in_sizes, int n_in,
                              void* d_out, int out_size, void* d_ws, size_t ws_size,
                              hipStream_t stream) {
    const float* x      = (const float*)d_in[0];      // [B, 4096] fp32
    const float* logits = (const float*)d_in[1];      // [B, 64]   fp32

    const int batch = in_sizes[1] / NUM_EXPERTS;      // 16384

    // d_out layout (concatenated, 4-byte elements):
    //   expanded_x:     [B*K, 4096] fp32
    //   expert_indices: [B*K]       int32 (raw bits into 4-byte slots)
    //   expert_weights: [B*K]       fp32
    float* out = (float*)d_out;
    const size_t expanded_elems = (size_t)batch * TOPK * ROW_FLOATS;
    int*   expert_idx = (int*)(out + expanded_elems);
    float* expert_w   = out + expanded_elems + (size_t)batch * TOPK;

    moe_router_kernel<<<(batch + 255) / 256, 256, 0, stream>>>(
        logits, expert_idx, expert_w, batch);

    moe_expand_kernel<<<batch, EXPAND_THREADS, 0, stream>>>(x, out);
}